// VTrace_52767968199014
// MI455X (gfx1250) — compile-verified
//
#include <hip/hip_runtime.h>
#include <hip/hip_bf16.h>

// V-trace (IMPALA) fused loss for MI455X / gfx1250.
// Bandwidth-bound streaming kernel: logits staged global->LDS with CDNA5
// async copies (ASYNCcnt, GVS saddr addressing, non-temporal hint), wave32
// shuffle reductions, deterministic 3-stage reduction (no FP atomics).

namespace {

constexpr int   Tt = 256;
constexpr int   Bb = 1024;
constexpr int   Nn = 64;
constexpr int   TB = Tt * Bb;
constexpr float GAMMA  = 0.99f;
constexpr float LAMBDA = 0.95f;

// k1 tiling
constexpr int ROWS_PER_BLOCK = 256;   // rows handled per block over 8 chunks
constexpr int CH             = 32;    // rows staged per chunk
constexpr int NCH            = ROWS_PER_BLOCK / CH;   // 8
constexpr int STRIDE         = 68;    // padded row stride in floats (16B aligned)
constexpr int STAGE_FLOATS   = CH * STRIDE;           // 2176
constexpr int LDS_BYTES      = 4 * STAGE_FLOATS * 4;  // T[2] + B[2] stages = 34816 B

// CDNA5 async copy, GVS mode: mem = SGPR_base + VGPR_offset; per-lane 16B -> LDS.
// Streams are read-once => non-temporal hint.
__device__ __forceinline__ void async_ld16(unsigned lds_byte_off, unsigned gvoff,
                                           const float* base) {
    asm volatile("global_load_async_to_lds_b128 %0, %1, %2 th:TH_LOAD_NT"
                 :: "v"(lds_byte_off), "v"(gvoff), "s"(base)
                 : "memory");
}
__device__ __forceinline__ void wait_async_le(int remaining) {
    if (remaining == 0) asm volatile("s_wait_asynccnt 0" ::: "memory");
    else                asm volatile("s_wait_asynccnt 4" ::: "memory");
}

__device__ __forceinline__ float red8_add(float v) {
    v += __shfl_xor(v, 1, 32);
    v += __shfl_xor(v, 2, 32);
    v += __shfl_xor(v, 4, 32);
    return v;
}
__device__ __forceinline__ float red8_max(float v) {
    v = fmaxf(v, __shfl_xor(v, 1, 32));
    v = fmaxf(v, __shfl_xor(v, 2, 32));
    v = fmaxf(v, __shfl_xor(v, 4, 32));
    return v;
}

__global__ __launch_bounds__(256)
void vtrace_rows_kernel(const float* __restrict__ tgt,
                        const float* __restrict__ beh,
                        const int*   __restrict__ act,
                        const float* __restrict__ value,
                        const float* __restrict__ reward,
                        float* __restrict__ fdelta,
                        float* __restrict__ fr,
                        float* __restrict__ flp,
                        float* __restrict__ fent) {
    extern __shared__ float smem[];   // dynamic LDS starts at offset 0 (no static LDS)
    const int tid       = threadIdx.x;
    const int blockRow0 = blockIdx.x * ROWS_PER_BLOCK;

    // Lane-constant async-copy offsets (hoisted out of the chunk loop).
    // chunk = CH rows * 16 float4 per row per tensor = 512 float4 / tensor,
    // 256 threads -> 2 per thread per tensor.
    const int      e0   = tid;
    const int      e1   = tid + 256;
    const unsigned gRow0 = (unsigned)(e0 >> 4) * (Nn * 4u) + (unsigned)(e0 & 15) * 16u;
    const unsigned gRow1 = (unsigned)(e1 >> 4) * (Nn * 4u) + (unsigned)(e1 & 15) * 16u;
    const unsigned lOff0 = (unsigned)(e0 >> 4) * (STRIDE * 4u) + (unsigned)(e0 & 15) * 16u;
    const unsigned lOff1 = (unsigned)(e1 >> 4) * (STRIDE * 4u) + (unsigned)(e1 & 15) * 16u;
    const unsigned blockByte = (unsigned)blockRow0 * (Nn * 4u);

    // prologue: stage chunk 0 into stage buffer 0
    {
        const unsigned cb = blockByte;
        async_ld16(lOff0, cb + gRow0, tgt);
        async_ld16(lOff1, cb + gRow1, tgt);
        async_ld16((unsigned)(2 * STAGE_FLOATS * 4) + lOff0, cb + gRow0, beh);
        async_ld16((unsigned)(2 * STAGE_FLOATS * 4) + lOff1, cb + gRow1, beh);
    }

    for (int c = 0; c < NCH; ++c) {
        const int st = c & 1;
        if (c + 1 < NCH) {
            const unsigned tb = (unsigned)(st ^ 1) * (STAGE_FLOATS * 4u);
            const unsigned bb = (unsigned)(2 * STAGE_FLOATS * 4) + tb;
            const unsigned cb = blockByte + (unsigned)(c + 1) * (CH * Nn * 4u);
            async_ld16(tb + lOff0, cb + gRow0, tgt);
            async_ld16(tb + lOff1, cb + gRow1, tgt);
            async_ld16(bb + lOff0, cb + gRow0, beh);
            async_ld16(bb + lOff1, cb + gRow1, beh);
        }
        wait_async_le((c + 1 < NCH) ? 4 : 0);   // chunk c complete (in-order ASYNCcnt)
        __syncthreads();

        // ---- compute on staged chunk c: 8 lanes per row, 8 logits each ----
        const int row_l = tid >> 3;
        const int oct   = tid & 7;
        const int n0    = oct * 8;
        const int g_row = blockRow0 + c * CH + row_l;

        const int tbF = st * STAGE_FLOATS;
        const int bbF = 2 * STAGE_FLOATS + st * STAGE_FLOATS;

        float tv[8], bv[8];
        {
            const float4* tp = reinterpret_cast<const float4*>(&smem[tbF + row_l * STRIDE + n0]);
            float4 a0 = tp[0], a1 = tp[1];
            tv[0]=a0.x; tv[1]=a0.y; tv[2]=a0.z; tv[3]=a0.w;
            tv[4]=a1.x; tv[5]=a1.y; tv[6]=a1.z; tv[7]=a1.w;
            const float4* bp = reinterpret_cast<const float4*>(&smem[bbF + row_l * STRIDE + n0]);
            float4 c0 = bp[0], c1 = bp[1];
            bv[0]=c0.x; bv[1]=c0.y; bv[2]=c0.z; bv[3]=c0.w;
            bv[4]=c1.x; bv[5]=c1.y; bv[6]=c1.z; bv[7]=c1.w;
        }
        const int a = act[g_row];

        // target: logsumexp, entropy pieces, action logit
        float mt = tv[0];
#pragma unroll
        for (int k = 1; k < 8; ++k) mt = fmaxf(mt, tv[k]);
        mt = red8_max(mt);
        float st_ = 0.f, sxt = 0.f, xat = 0.f;
#pragma unroll
        for (int k = 0; k < 8; ++k) {
            const float e = __expf(tv[k] - mt);
            st_ += e;
            sxt += e * tv[k];
            if (n0 + k == a) xat += tv[k];
        }
        st_ = red8_add(st_); sxt = red8_add(sxt); xat = red8_add(xat);

        // behaviour: logsumexp + action logit
        float mb = bv[0];
#pragma unroll
        for (int k = 1; k < 8; ++k) mb = fmaxf(mb, bv[k]);
        mb = red8_max(mb);
        float sb = 0.f, xab = 0.f;
#pragma unroll
        for (int k = 0; k < 8; ++k) {
            sb += __expf(bv[k] - mb);
            if (n0 + k == a) xab += bv[k];
        }
        sb = red8_add(sb); xab = red8_add(xab);

        if (oct == 0) {
            const float lse_t = mt + __logf(st_);
            const float ent   = lse_t - sxt / st_;          // -sum p*logp
            const float lpta  = xat - lse_t;
            const float lse_b = mb + __logf(sb);
            const float lpba  = xab - lse_b;
            const float IS    = __expf(lpta - lpba);
            const float r     = fminf(IS, 1.0f);            // rho == c == pg_rho (clips all 1.0)
            const float v0 = value[g_row];
            const float v1 = value[g_row + Bb];
            const float rw = reward[g_row];
            fdelta[g_row] = r * (rw + GAMMA * v1 - v0);
            fr[g_row]     = r;
            flp[g_row]    = lpta;
            fent[g_row]   = ent;
        }
        __syncthreads();   // chunk buffer may be overwritten next iteration
    }
}

__global__ __launch_bounds__(32)
void vtrace_scan_kernel(const float* __restrict__ fdelta,
                        const float* __restrict__ fr,
                        const float* __restrict__ flp,
                        const float* __restrict__ fent,
                        const float* __restrict__ reward,
                        const float* __restrict__ value,
                        const float* __restrict__ weight,
                        float* __restrict__ partials) {
    const int b = blockIdx.x * 32 + threadIdx.x;   // one column per lane
    const float f = GAMMA * LAMBDA;
    float carry = 0.f;
    float ret_next = value[Tt * Bb + b];           // value[T]
    float pg = 0.f, vl = 0.f, el = 0.f;
    for (int t = Tt - 1; t >= 0; --t) {
        const int i = t * Bb + b;
        const float d  = fdelta[i];
        const float r  = fr[i];
        const float lp = flp[i];
        const float en = fent[i];
        const float rw = reward[i];
        const float v0 = value[i];
        const float w  = weight[i];
        carry = d + f * r * carry;                          // items_t
        const float adv = r * (rw + GAMMA * ret_next - v0); // pg_rhos * (...)
        pg += lp * adv * w;
        vl += carry * carry * w;                            // (returns - value)^2
        el += en * w;
        ret_next = v0 + carry;                              // returns_t
    }
#pragma unroll
    for (int m = 1; m < 32; m <<= 1) {
        pg += __shfl_xor(pg, m, 32);
        vl += __shfl_xor(vl, m, 32);
        el += __shfl_xor(el, m, 32);
    }
    if (threadIdx.x == 0) {
        partials[blockIdx.x * 3 + 0] = pg;
        partials[blockIdx.x * 3 + 1] = vl;
        partials[blockIdx.x * 3 + 2] = el;
    }
}

__global__ __launch_bounds__(32)
void vtrace_final_kernel(const float* __restrict__ partials, float* __restrict__ out) {
    const int j = threadIdx.x;          // 32 lanes <-> 32 partial triples
    float pg = partials[j * 3 + 0];
    float vl = partials[j * 3 + 1];
    float el = partials[j * 3 + 2];
#pragma unroll
    for (int m = 1; m < 32; m <<= 1) {  // fixed tree -> deterministic
        pg += __shfl_xor(pg, m, 32);
        vl += __shfl_xor(vl, m, 32);
        el += __shfl_xor(el, m, 32);
    }
    if (j == 0) {
        const float inv = 1.0f / (float)TB;
        out[0] = -pg * inv;   // pg_loss
        out[1] =  vl * inv;   // value_loss
        out[2] =  el * inv;   // entropy_loss
    }
}

} // namespace

extern "C" void kernel_launch(void* const* d_in, const int* in_sizes, int n_in,
                              void* d_out, int out_size, void* d_ws, size_t ws_size,
                              hipStream_t stream) {
    (void)in_sizes; (void)n_in; (void)out_size; (void)ws_size;
    const float* tgt    = (const float*)d_in[0];   // target_output (T,B,N)
    const float* beh    = (const float*)d_in[1];   // behaviour_output (T,B,N)
    const int*   act    = (const int*)  d_in[2];   // action (T,B)
    const float* value  = (const float*)d_in[3];   // value (T+1,B)
    const float* reward = (const float*)d_in[4];   // reward (T,B)
    const float* weight = (const float*)d_in[5];   // weight (T,B)

    float* ws      = (float*)d_ws;
    float* fdelta  = ws;
    float* fr      = ws + (size_t)TB;
    float* flp     = ws + (size_t)2 * TB;
    float* fent    = ws + (size_t)3 * TB;
    float* partial = ws + (size_t)4 * TB;          // 32 * 3 floats

    vtrace_rows_kernel<<<TB / ROWS_PER_BLOCK, 256, LDS_BYTES, stream>>>(
        tgt, beh, act, value, reward, fdelta, fr, flp, fent);
    vtrace_scan_kernel<<<Bb / 32, 32, 0, stream>>>(
        fdelta, fr, flp, fent, reward, value, weight, partial);
    vtrace_final_kernel<<<1, 32, 0, stream>>>(partial, (float*)d_out);
}